// ConditionalPointerHead_14998025797666
// MI455X (gfx1250) — compile-verified
//
#include <hip/hip_runtime.h>

#define BB   16
#define LL   4096
#define DD   1024
#define KP   16          // top-K padded to 16 (real K=5, pad weights = 0)
#define NEG  (-1e9f)
#define SCALE (0.03125f) // 1/sqrt(1024)

typedef __attribute__((ext_vector_type(2))) float v2f;
typedef __attribute__((ext_vector_type(8))) float v8f;

#ifndef __has_builtin
#define __has_builtin(x) 0
#endif
#if __has_builtin(__builtin_amdgcn_global_load_async_to_lds_b128)
#define ASYNC_LDS 1
#else
#define ASYNC_LDS 0
#endif

// 128-bit transfer granule, matching the builtin's parameter type
typedef int b128_t __attribute__((vector_size(16)));
typedef __attribute__((address_space(1))) b128_t* gb128_p;  // global
typedef __attribute__((address_space(3))) b128_t* lb128_p;  // LDS

#define GPTR(p) ((gb128_p)(unsigned long long)(uintptr_t)(p))
#define LPTR(p) ((lb128_p)(unsigned int)(uintptr_t)(p))

__device__ __forceinline__ void wait_async0() {
#if __has_builtin(__builtin_amdgcn_s_wait_asynccnt)
  __builtin_amdgcn_s_wait_asynccnt(0);
#else
  asm volatile("s_wait_asynccnt 0x0" ::: "memory");
#endif
}

// Contiguous global -> LDS stage (16B granules). Caller must __syncthreads().
__device__ __forceinline__ void stage_lds(float* __restrict__ dst,
                                          const float* __restrict__ src,
                                          int nfloats, int t) {
#if ASYNC_LDS
  for (int i = t; i < (nfloats >> 2); i += 256)
    __builtin_amdgcn_global_load_async_to_lds_b128(GPTR(src + 4 * i),
                                                   LPTR(dst + 4 * i), 0, 0);
  wait_async0();
#else
  for (int i = t; i < (nfloats >> 2); i += 256)
    ((float4*)dst)[i] = ((const float4*)src)[i];
#endif
}

__device__ __forceinline__ v8f wmma4(v2f a, v2f b, v8f c) {
  // D(16x16,f32) = A(16x4,f32) * B(4x16,f32) + C
  return __builtin_amdgcn_wmma_f32_16x16x4_f32(false, a, false, b, (short)0, c,
                                               false, false);
}

// ---------------- 1) start_logits = H @ w_start + b_start, masked ----------
__global__ void k_start(const float* __restrict__ H, const int* __restrict__ msk,
                        const float* __restrict__ wst, const float* __restrict__ bst,
                        float* __restrict__ out) {
  const int row  = blockIdx.x * 8 + (threadIdx.x >> 5);   // 8 wave32 per block
  const int lane = threadIdx.x & 31;
  const float* hr = H + (size_t)row * DD;
  float s = 0.f;
  for (int j = lane; j < DD; j += 32) s = fmaf(hr[j], wst[j], s);
  for (int o = 16; o; o >>= 1) s += __shfl_down(s, o, 32);
  if (lane == 0) {
    float v = s + bst[0];
    if (msk[row] == 0) v = NEG;
    out[row] = v;
  }
}

// ---------------- 2) softmax stats + top-5 per batch -----------------------
__global__ void k_topk(const float* __restrict__ logits,
                       float* __restrict__ tw, int* __restrict__ tidx) {
  const int b = blockIdx.x, t = threadIdx.x;
  __shared__ float red[256];
  __shared__ int   redi[256];
  __shared__ float s_max, s_sum, cp[5];
  __shared__ int   chosen[5];
  const float* row = logits + (size_t)b * LL;

  float mx = -3.0e38f;
  for (int l = t; l < LL; l += 256) mx = fmaxf(mx, row[l]);
  red[t] = mx; __syncthreads();
  for (int s = 128; s > 0; s >>= 1) { if (t < s) red[t] = fmaxf(red[t], red[t + s]); __syncthreads(); }
  if (t == 0) s_max = red[0];
  __syncthreads();
  mx = s_max;

  float sm = 0.f;
  for (int l = t; l < LL; l += 256) sm += __expf(row[l] - mx);
  red[t] = sm; __syncthreads();
  for (int s = 128; s > 0; s >>= 1) { if (t < s) red[t] += red[t + s]; __syncthreads(); }
  if (t == 0) s_sum = red[0];
  __syncthreads();

  for (int it = 0; it < 5; ++it) {
    float bv = -3.0e38f; int bi = LL;
    for (int l = t; l < LL; l += 256) {
      bool skip = false;
      for (int j = 0; j < it; ++j) skip = skip || (chosen[j] == l);
      if (skip) continue;
      float v = row[l];
      if (v > bv || (v == bv && l < bi)) { bv = v; bi = l; }
    }
    red[t] = bv; redi[t] = bi; __syncthreads();
    for (int s = 128; s > 0; s >>= 1) {
      if (t < s) {
        if (red[t + s] > red[t] || (red[t + s] == red[t] && redi[t + s] < redi[t])) {
          red[t] = red[t + s]; redi[t] = redi[t + s];
        }
      }
      __syncthreads();
    }
    if (t == 0) { chosen[it] = redi[0]; cp[it] = __expf(red[0] - s_max) / s_sum; }
    __syncthreads();
  }
  if (t == 0) {
    float sw = 1e-9f;
    for (int k = 0; k < 5; ++k) sw += cp[k];
    for (int k = 0; k < 5; ++k) { tw[b * KP + k] = cp[k] / sw; tidx[b * KP + k] = chosen[k]; }
    for (int k = 5; k < KP; ++k) { tw[b * KP + k] = 0.f; tidx[b * KP + k] = 0; }
  }
}

// ---------------- 3/8) row @ W + bias  (Q proj / V proj) -------------------
__global__ void k_rowproj(const float* __restrict__ rows, const float* __restrict__ W,
                          const float* __restrict__ bias, float* __restrict__ out,
                          const int* __restrict__ gidx, const float* __restrict__ Hfull) {
  const int b = blockIdx.y, k = blockIdx.x, t = threadIdx.x;
  __shared__ float hr[DD];
  const float* src;
  if (gidx) { int idx = gidx[b * KP + k]; src = Hfull + ((size_t)b * LL + idx) * DD; }
  else        src = rows + (size_t)(b * KP + k) * DD;
  for (int d = t; d < DD; d += 256) hr[d] = src[d];
  __syncthreads();
  float acc[4];
#pragma unroll
  for (int c = 0; c < 4; ++c) acc[c] = bias[t + 256 * c];
  for (int d = 0; d < DD; ++d) {
    const float h = hr[d];
    const float* wr = W + (size_t)d * DD;
#pragma unroll
    for (int c = 0; c < 4; ++c) acc[c] = fmaf(h, wr[t + 256 * c], acc[c]);
  }
#pragma unroll
  for (int c = 0; c < 4; ++c) out[(size_t)(b * KP + k) * DD + t + 256 * c] = acc[c];
}

// ------- 4/9) X @ W^T, output TRANSPOSED [d][k] (async-stage friendly) -----
__global__ void k_projT(const float* __restrict__ X, const float* __restrict__ W,
                        float* __restrict__ outT) {
  const int b = blockIdx.y, t = threadIdx.x;
  const int d = blockIdx.x * 256 + t;
  __shared__ float xs[KP * DD];   // 64 KB (320 KB/WGP on CDNA5)
  for (int idx = t; idx < KP * DD; idx += 256)
    xs[idx] = X[(size_t)b * KP * DD + idx];
  __syncthreads();
  float acc[KP];
#pragma unroll
  for (int k = 0; k < KP; ++k) acc[k] = 0.f;
  const float* wr = W + (size_t)d * DD;
  for (int dd = 0; dd < DD; ++dd) {
    const float wv = wr[dd];            // per-lane row stream
#pragma unroll
    for (int k = 0; k < KP; ++k) acc[k] = fmaf(xs[k * DD + dd], wv, acc[k]); // LDS broadcast
  }
#pragma unroll
  for (int k = 0; k < KP; ++k)
    outT[(size_t)b * KP * DD + d * KP + k] = acc[k];
}

// ---------------- 5) WMMA score pass: scores = scale * Qk . H^T ------------
__global__ void k_scores(const float* __restrict__ H, const float* __restrict__ QkT,
                         const int* __restrict__ msk, float* __restrict__ sc) {
  const int b = blockIdx.y, t = threadIdx.x;
  const int wave = t >> 5, lane = t & 31;
  const int m = lane & 15, hi = lane >> 4, c0 = 2 * hi;
  const int l0 = blockIdx.x * 128 + wave * 16;
  __shared__ __align__(16) float qs[KP * DD];  // A operand [d][k], conflict-free
  stage_lds(qs, QkT + (size_t)b * KP * DD, KP * DD, t);
  __syncthreads();
  const float* hrow = H + ((size_t)(b * LL + l0 + m)) * DD;
  v8f acc = {};
  for (int kc = 0; kc < DD; kc += 256) {
    __builtin_prefetch(hrow + kc + 256, 0, 3);     // WGP-scope global_prefetch_b8
    for (int k0 = kc; k0 < kc + 256; k0 += 4) {
      v2f a, bb;
      a.x  = qs[(k0 + c0) * KP + m];
      a.y  = qs[(k0 + c0 + 1) * KP + m];
      bb.x = hrow[k0 + c0];
      bb.y = hrow[k0 + c0 + 1];
      acc = wmma4(a, bb, acc);
    }
  }
  const int n = m;
  const int pad = (msk[b * LL + l0 + n] == 0);
#pragma unroll
  for (int r = 0; r < 8; ++r) {
    const int M = r + 8 * hi;
    float v = acc[r] * SCALE;
    if (pad) v = NEG;
    sc[(size_t)(b * KP + M) * LL + l0 + n] = v;
  }
}

// ------- 6) row softmax over L; also emit transposed copy [b][l][k] --------
__global__ void k_softmax(float* __restrict__ sc, float* __restrict__ aT) {
  const int kk = blockIdx.x, b = blockIdx.y, t = threadIdx.x;
  const int row = b * KP + kk;
  float* s = sc + (size_t)row * LL;
  __shared__ float red[256];
  __shared__ float sm, ss;
  float mx = -3.0e38f;
  for (int l = t; l < LL; l += 256) mx = fmaxf(mx, s[l]);
  red[t] = mx; __syncthreads();
  for (int q = 128; q > 0; q >>= 1) { if (t < q) red[t] = fmaxf(red[t], red[t + q]); __syncthreads(); }
  if (t == 0) sm = red[0];
  __syncthreads();
  float sum = 0.f;
  for (int l = t; l < LL; l += 256) sum += __expf(s[l] - sm);
  red[t] = sum; __syncthreads();
  for (int q = 128; q > 0; q >>= 1) { if (t < q) red[t] += red[t + q]; __syncthreads(); }
  if (t == 0) ss = red[0];
  __syncthreads();
  const float inv = 1.0f / ss;
  for (int l = t; l < LL; l += 256) {
    const float v = __expf(s[l] - sm) * inv;
    s[l] = v;
    aT[((size_t)b * LL + l) * KP + kk] = v;
  }
}

// ---------------- 7) WMMA ctx pass: ctxH = attn . H ------------------------
__global__ void k_ctx(const float* __restrict__ H, const float* __restrict__ attnT,
                      float* __restrict__ ctxH) {
  const int b = blockIdx.y, t = threadIdx.x;
  const int wave = t >> 5, lane = t & 31;
  const int m = lane & 15, hi = lane >> 4, c0 = 2 * hi;
  const int dcol = blockIdx.x * 128 + wave * 16 + m;
  __shared__ __align__(16) float at[256 * KP];  // attn chunk [l][k]
  v8f acc = {};
  for (int lch = 0; lch < LL; lch += 256) {
    stage_lds(at, attnT + ((size_t)b * LL + lch) * KP, 256 * KP, t);
    __syncthreads();
    for (int kk = 0; kk < 256; kk += 4) {
      v2f a, bb;
      a.x = at[(kk + c0) * KP + m];
      a.y = at[(kk + c0 + 1) * KP + m];
      const size_t rb = ((size_t)(b * LL + lch + kk + c0)) * DD + dcol;
      __builtin_prefetch(&H[rb + 64 * (size_t)DD], 0, 3);  // 64 rows ahead
      bb.x = H[rb];
      bb.y = H[rb + DD];
      acc = wmma4(a, bb, acc);
    }
    __syncthreads();
  }
#pragma unroll
  for (int r = 0; r < 8; ++r)
    ctxH[(size_t)(b * KP + r + 8 * hi) * DD + dcol] = acc[r];
}

// ---------------- 10) cb = cond_ctx . b_cmp --------------------------------
__global__ void k_cb(const float* __restrict__ cc, const float* __restrict__ bc,
                     float* __restrict__ cb) {
  const int row = blockIdx.y * KP + blockIdx.x, t = threadIdx.x;
  __shared__ float red[256];
  float s = 0.f;
  for (int d = t; d < DD; d += 256) s = fmaf(cc[(size_t)row * DD + d], bc[d], s);
  red[t] = s; __syncthreads();
  for (int q = 128; q > 0; q >>= 1) { if (t < q) red[t] += red[t + q]; __syncthreads(); }
  if (t == 0) cb[row] = red[0];
}

// ---------------- 11) WMMA end pass + weighted K-reduction -----------------
__global__ void k_end(const float* __restrict__ H, const float* __restrict__ C2T,
                      const float* __restrict__ cb, const float* __restrict__ w,
                      const int* __restrict__ msk, float* __restrict__ out) {
  const int b = blockIdx.y, t = threadIdx.x;
  const int wave = t >> 5, lane = t & 31;
  const int m = lane & 15, hi = lane >> 4, c0 = 2 * hi;
  const int l0 = blockIdx.x * 128 + wave * 16;
  __shared__ __align__(16) float qs[KP * DD];
  stage_lds(qs, C2T + (size_t)b * KP * DD, KP * DD, t);
  __syncthreads();
  const float* hrow = H + ((size_t)(b * LL + l0 + m)) * DD;
  v8f acc = {};
  for (int kc = 0; kc < DD; kc += 256) {
    __builtin_prefetch(hrow + kc + 256, 0, 3);
    for (int k0 = kc; k0 < kc + 256; k0 += 4) {
      v2f a, bb;
      a.x  = qs[(k0 + c0) * KP + m];
      a.y  = qs[(k0 + c0 + 1) * KP + m];
      bb.x = hrow[k0 + c0];
      bb.y = hrow[k0 + c0 + 1];
      acc = wmma4(a, bb, acc);
    }
  }
  float partial = 0.f;
#pragma unroll
  for (int r = 0; r < 8; ++r) {
    const int M = r + 8 * hi;
    partial = fmaf(w[b * KP + M], SCALE * (acc[r] + cb[b * KP + M]), partial);
  }
  const float tot = partial + __shfl_xor(partial, 16, 32);
  const int n = m;
  if (hi == 0) {
    float v = (msk[b * LL + l0 + n] == 0) ? NEG : tot;
    out[(size_t)b * LL + l0 + n] = v;
  }
}

extern "C" void kernel_launch(void* const* d_in, const int* in_sizes, int n_in,
                              void* d_out, int out_size, void* d_ws, size_t ws_size,
                              hipStream_t stream) {
  (void)in_sizes; (void)n_in; (void)out_size; (void)ws_size;
  const float* H     = (const float*)d_in[0];
  const int*   amask = (const int*)d_in[1];
  const float* w_st  = (const float*)d_in[2];
  const float* b_st  = (const float*)d_in[3];
  const float* w_q   = (const float*)d_in[4];
  const float* b_q   = (const float*)d_in[5];
  const float* w_k   = (const float*)d_in[6];
  /* b_k cancels in the attention softmax (per-row constant) */
  const float* w_v   = (const float*)d_in[8];
  const float* b_v   = (const float*)d_in[9];
  const float* w_cmp = (const float*)d_in[10];
  const float* b_cmp = (const float*)d_in[11];

  float* start_logits = (float*)d_out;                    // [B,L]
  float* end_logits   = start_logits + (size_t)BB * LL;   // [B,L]

  float* ws    = (float*)d_ws;                            // ~13.6 MB used
  float* tw    = ws;                                      // [B,KP]
  int*   tidx  = (int*)(ws + 256);                        // [B,KP]
  float* Q     = ws + 512;                                // [B,KP,D]
  float* QkT   = Q + (size_t)BB * KP * DD;                // [B,D,KP] (transposed)
  float* sc    = QkT + (size_t)BB * KP * DD;              // [B,KP,L]
  float* aT    = sc + (size_t)BB * KP * LL;               // [B,L,KP] (transposed)
  float* ctxH  = aT + (size_t)BB * KP * LL;               // [B,KP,D]
  float* cctx  = ctxH + (size_t)BB * KP * DD;             // [B,KP,D]
  float* C2T   = cctx + (size_t)BB * KP * DD;             // [B,D,KP] (transposed)
  float* cb    = C2T + (size_t)BB * KP * DD;              // [B,KP]

  k_start  <<<BB * LL / 8, 256, 0, stream>>>(H, amask, w_st, b_st, start_logits);
  k_topk   <<<BB, 256, 0, stream>>>(start_logits, tw, tidx);
  k_rowproj<<<dim3(KP, BB), 256, 0, stream>>>(nullptr, w_q, b_q, Q, tidx, H);
  k_projT  <<<dim3(4, BB), 256, 0, stream>>>(Q, w_k, QkT);
  k_scores <<<dim3(LL / 128, BB), 256, 0, stream>>>(H, QkT, amask, sc);
  k_softmax<<<dim3(KP, BB), 256, 0, stream>>>(sc, aT);
  k_ctx    <<<dim3(DD / 128, BB), 256, 0, stream>>>(H, aT, ctxH);
  k_rowproj<<<dim3(KP, BB), 256, 0, stream>>>(ctxH, w_v, b_v, cctx, nullptr, nullptr);
  k_projT  <<<dim3(4, BB), 256, 0, stream>>>(cctx, w_cmp, C2T);
  k_cb     <<<dim3(KP, BB), 256, 0, stream>>>(cctx, b_cmp, cb);
  k_end    <<<dim3(LL / 128, BB), 256, 0, stream>>>(H, C2T, cb, tw, amask, end_logits);
}